// CrossTransformer_44169443672538
// MI455X (gfx1250) — compile-verified
//
#include <hip/hip_runtime.h>
#include <hip/hip_bf16.h>
#include <math.h>

typedef __attribute__((ext_vector_type(2))) float v2f;
typedef __attribute__((ext_vector_type(8))) float v8f;
typedef __attribute__((ext_vector_type(4))) unsigned int v4u;
typedef __attribute__((ext_vector_type(8))) int v8i;
typedef __attribute__((ext_vector_type(4))) int v4i;

#define BSZ 2
#define LLDIM 255
#define WWDIM 255
#define DDIM 128
#define NROW (BSZ * LLDIM)              // 510 attention rows
#define NTOK (NROW * WWDIM)             // 130050 tokens / pixels
#define MTILES ((NTOK + 15) / 16)       // 8129 M-tiles

#if __has_builtin(__builtin_amdgcn_tensor_load_to_lds)
#define HAVE_TDM 1
#else
#define HAVE_TDM 0
#endif

static __device__ __forceinline__ v8f wmma4(v2f a, v2f b, v8f c) {
  // D = A(16x4,f32) * B(4x16,f32) + C ; full fp32 on the matrix pipe
  return __builtin_amdgcn_wmma_f32_16x16x4_f32(false, a, false, b, (short)0, c,
                                               false, false);
}

#if HAVE_TDM
// Tensor Data Mover: 2D tile (dim0 contiguous elems x dim1 rows, f32) -> LDS.
// D# layout per CDNA5 ISA.
static __device__ __forceinline__ void tdm_load_2d(unsigned int lds_off,
                                                   const void* gptr,
                                                   unsigned int dim0,
                                                   unsigned int dim1,
                                                   unsigned int stride0) {
  unsigned long long ga = (unsigned long long)gptr;
  v4u g0;
  g0[0] = 1u;                                   // count=1, user descriptor
  g0[1] = lds_off;                              // lds_addr (bytes)
  g0[2] = (unsigned int)(ga & 0xffffffffu);     // global_addr[31:0]
  g0[3] = (unsigned int)((ga >> 32) & 0x1ffffffu) | (2u << 30);  // addr[56:32], type=2
  v8i g1;
  g1[0] = (int)(2u << 16);                      // data_size=4B; no mask/pad/iter
  g1[1] = (int)((dim0 & 0xffffu) << 16);        // tensor_dim0[15:0] @ bits 63:48
  g1[2] = (int)(((dim0 >> 16) & 0xffffu) | ((dim1 & 0xffffu) << 16));
  g1[3] = (int)(((dim1 >> 16) & 0xffffu) | ((dim0 & 0xffffu) << 16));  // tile_dim0
  g1[4] = (int)(dim1 & 0xffffu);                // tile_dim1; tile_dim2=0
  g1[5] = (int)stride0;                         // tensor_dim0_stride[31:0]
  g1[6] = 0;
  g1[7] = 0;
  v4i z = {0, 0, 0, 0};
#if defined(__clang_major__) && (__clang_major__ >= 23)
  v8i z8 = {0, 0, 0, 0, 0, 0, 0, 0};
  __builtin_amdgcn_tensor_load_to_lds(g0, g1, z, z, z8, 0);
#else
  __builtin_amdgcn_tensor_load_to_lds(g0, g1, z, z, 0);
#endif
}
#endif

// ---------------- elementwise prep: clip + positional embedding ----------
__global__ void k_prep(const float* __restrict__ x, const float* __restrict__ pe,
                       float* __restrict__ xp) {
  size_t idx = (size_t)blockIdx.x * blockDim.x + threadIdx.x;
  if (idx >= (size_t)NTOK * DDIM) return;
  int c = (int)(idx & (DDIM - 1));
  int j = (int)((idx >> 7) % WWDIM);
  float v = x[idx];
  v = fminf(fmaxf(v, -1000.f), 1000.f);
  v += pe[(j < (WWDIM / 2) ? 0 : 1) * DDIM + c];
  xp[idx] = v;
}

// ---------------- row_valid: row is valid if not fully masked ------------
__global__ void k_rowvalid(const int* __restrict__ mask, int* __restrict__ rv) {
  int warp = (int)(((size_t)blockIdx.x * blockDim.x + threadIdx.x) >> 5);
  int lane = threadIdx.x & 31;
  if (warp >= NROW) return;
  int s = 0;
  for (int c = lane; c < WWDIM; c += 32) s += mask[warp * WWDIM + c];
  for (int o = 16; o; o >>= 1) s += __shfl_xor(s, o);
  if (lane == 0) rv[warp] = (s != WWDIM) ? 1 : 0;
}

// ---------------- conv weight repack: OIHW -> [co][tap][ci] --------------
__global__ void k_repack(const float* __restrict__ w, float* __restrict__ wt) {
  int idx = blockIdx.x * blockDim.x + threadIdx.x;
  if (idx >= DDIM * DDIM * 9) return;
  int ci = idx & (DDIM - 1);
  int tap = (idx >> 7) % 9;
  int co = idx / (9 * DDIM);
  wt[idx] = w[(co * DDIM + ci) * 9 + tap];
}

// ---------------- QKV GEMM: (NTOK,128) @ W^T(128,384) + b, clip ----------
// gather=1 pulls the vertical-branch diagonal rows of x_proc on the fly.
__global__ void k_qkv(const float* __restrict__ xp, const float* __restrict__ W,
                      const float* __restrict__ bias, float* __restrict__ qkv,
                      int gather) {
  int wave = threadIdx.x >> 5, lane = threadIdx.x & 31;
  int lm = lane & 15, lk = (lane >> 4) * 2;
  int mbase = blockIdx.x * 16;
  int nbase = (blockIdx.y * 8 + wave) * 16;
  int m = mbase + lm;
  int mr = (m < NTOK) ? m : NTOK - 1;
  const float* arow;
  if (!gather) {
    arow = xp + (size_t)mr * DDIM;
  } else {
    int b = mr / (LLDIM * WWDIM);
    int rem = mr % (LLDIM * WWDIM);
    int i = rem / WWDIM, j = rem % WWDIM;
    int cc = WWDIM - 1 - j;
    int si = (i - cc + WWDIM / 2 + WWDIM) % WWDIM;
    arow = xp + (((size_t)(b * LLDIM + si)) * WWDIM + cc) * (size_t)DDIM;
  }
  int n = nbase + lm;
  const float* brow = W + (size_t)n * DDIM;
  v8f acc = {};
  for (int k0 = 0; k0 < DDIM; k0 += 4) {
    if (k0 + 32 < DDIM) __builtin_prefetch(arow + k0 + 32, 0, 1);
    v2f a, b2;
    a.x = arow[k0 + lk];  a.y = arow[k0 + lk + 1];
    b2.x = brow[k0 + lk]; b2.y = brow[k0 + lk + 1];
    acc = wmma4(a, b2, acc);
  }
  float bn = bias[n];
  for (int g = 0; g < 8; ++g) {
    int mm = mbase + g + ((lane >> 4) << 3);
    if (mm < NTOK) {
      float v = acc[g] + bn;
      v = fminf(fmaxf(v, -10000.f), 10000.f);
      qkv[(size_t)mm * 384 + n] = v;
    }
  }
}

// ---------------- rotary (in-place on q and k parts of qkv) --------------
__global__ void k_rotary(float* __restrict__ qkv) {
  size_t idx = (size_t)blockIdx.x * blockDim.x + threadIdx.x;
  if (idx >= (size_t)NTOK * 64) return;
  int c = (int)(idx & 63);
  int tok = (int)(idx >> 6);
  int j = tok % WWDIM;
  double freq = pow(10000.0, -(double)(2 * c) / (double)DDIM);
  double ang = (double)j * freq;
  float sn = (float)sin(ang), co = (float)cos(ang);
  float* base = qkv + (size_t)tok * 384;
  float x1 = base[2 * c], x2 = base[2 * c + 1];
  base[2 * c] = x1 * co - x2 * sn;
  base[2 * c + 1] = x2 * co + x1 * sn;
  x1 = base[DDIM + 2 * c]; x2 = base[DDIM + 2 * c + 1];
  base[DDIM + 2 * c] = x1 * co - x2 * sn;
  base[DDIM + 2 * c + 1] = x2 * co + x1 * sn;
}

// ---------------- attention: one workgroup per row, 256KB LDS scores -----
__global__ void k_attn(const float* __restrict__ qkv, const int* __restrict__ mask,
                       float* __restrict__ out) {
  extern __shared__ float sm[];  // 256 x 256 f32 score matrix
  int r = blockIdx.x;
  int wave = threadIdx.x >> 5, lane = threadIdx.x & 31;
  int lm = lane & 15, lk = (lane >> 4) * 2;
  const float* base = qkv + (size_t)r * WWDIM * 384;
  const int* mrow = mask + (size_t)r * WWDIM;
  const float inv_scale = 0.08838834764831845f;  // 1/sqrt(128)
  int mbase = wave * 16;

  // ---- phase 1: S = clip(Q K^T / scale) + (-1e4 * mask) ----
  {
    int qm = mbase + lm; if (qm > WWDIM - 1) qm = WWDIM - 1;
    const float* qrow = base + (size_t)qm * 384;
    for (int nt = 0; nt < 16; ++nt) {
      int nbase = nt * 16;
      int kn = nbase + lm; if (kn > WWDIM - 1) kn = WWDIM - 1;
      const float* krow = base + (size_t)kn * 384 + DDIM;
      v8f acc = {};
      for (int k0 = 0; k0 < DDIM; k0 += 4) {
        v2f a, b;
        a.x = qrow[k0 + lk]; a.y = qrow[k0 + lk + 1];
        b.x = krow[k0 + lk]; b.y = krow[k0 + lk + 1];
        acc = wmma4(a, b, acc);
      }
      for (int g = 0; g < 8; ++g) {
        int mm = mbase + g + ((lane >> 4) << 3);
        int nn = nbase + lm;
        float s = acc[g] * inv_scale;
        s = fminf(fmaxf(s, -10000.f), 10000.f);
        float mf = (nn < WWDIM) ? (float)mrow[nn] : 1.f;
        sm[mm * 256 + nn] = s - 10000.f * mf;
      }
    }
  }
  __syncthreads();

  // ---- phase 2: wave-per-row softmax + post-mask zeroing ----
  for (int row = wave; row < WWDIM; row += 16) {
    float mx = -3.4e38f;
    for (int c = lane; c < WWDIM; c += 32) mx = fmaxf(mx, sm[row * 256 + c]);
    for (int o = 16; o; o >>= 1) mx = fmaxf(mx, __shfl_xor(mx, o));
    float sum = 0.f;
    for (int c = lane; c < WWDIM; c += 32) {
      float e = __expf(sm[row * 256 + c] - mx);
      sm[row * 256 + c] = e; sum += e;
    }
    for (int o = 16; o; o >>= 1) sum += __shfl_xor(sum, o);
    float inv = 1.f / sum;
    for (int c = lane; c < WWDIM; c += 32) {
      float v = sm[row * 256 + c] * inv;
      if (mrow[c] != 0) v = 0.f;
      sm[row * 256 + c] = v;
    }
    if (lane == 0) sm[row * 256 + 255] = 0.f;  // pad column contributes zero
  }
  __syncthreads();

  // ---- phase 3: O = A(LDS,255x256) @ V(256x128) ----
  const v2f* arow2 = (const v2f*)(sm + (mbase + lm) * 256);  // 8B-aligned pairs
  for (int nt = 0; nt < 8; ++nt) {
    int nbase = nt * 16;
    v8f acc = {};
    for (int k0 = 0; k0 < 256; k0 += 4) {
      v2f a = arow2[(k0 + lk) >> 1];   // ds_load_b64 from score matrix
      v2f b;
      int kr0 = k0 + lk;     if (kr0 > WWDIM - 1) kr0 = WWDIM - 1;
      int kr1 = k0 + lk + 1; if (kr1 > WWDIM - 1) kr1 = WWDIM - 1;
      b.x = base[(size_t)kr0 * 384 + 256 + nbase + lm];
      b.y = base[(size_t)kr1 * 384 + 256 + nbase + lm];
      acc = wmma4(a, b, acc);
    }
    for (int g = 0; g < 8; ++g) {
      int mm = mbase + g + ((lane >> 4) << 3);
      if (mm < WWDIM) out[((size_t)r * WWDIM + mm) * DDIM + nbase + lm] = acc[g];
    }
  }
}

// ---------------- dense: concat(hv,vv)(NTOK,256) @ W^T + b, rv zeroing ---
// Weights (128x256 f32, 131KB) staged in LDS once per block via the Tensor
// Data Mover and shared by all 8 waves.
__global__ void k_dense(const float* __restrict__ hv, const float* __restrict__ vv,
                        const float* __restrict__ W, const float* __restrict__ bias,
                        const int* __restrict__ rv, float* __restrict__ outb) {
  extern __shared__ float wsm[];  // 128 x 256 dense weights
#if HAVE_TDM
  if (threadIdx.x < 32) {  // wave 0 issues the DMA; TENSORcnt is per-wave
    tdm_load_2d((unsigned int)(uintptr_t)wsm, (const void*)W, 256u, 128u, 256u);
    __builtin_amdgcn_s_wait_tensorcnt(0);
  }
#else
  for (int t = threadIdx.x; t < DDIM * 256 / 4; t += blockDim.x)
    ((float4*)wsm)[t] = ((const float4*)W)[t];
#endif
  __syncthreads();

  int wave = threadIdx.x >> 5, lane = threadIdx.x & 31;
  int lm = lane & 15, lk = (lane >> 4) * 2;
  int mbase = blockIdx.x * 16;
  int nbase = wave * 16;
  int m = mbase + lm;
  int mr = (m < NTOK) ? m : NTOK - 1;
  const float* h = hv + (size_t)mr * DDIM;
  const float* vr = vv + (size_t)mr * DDIM;
  int n = nbase + lm;
  const v2f* brow2 = (const v2f*)(wsm + n * 256);
  v8f acc = {};
  for (int k0 = 0; k0 < DDIM; k0 += 4) {
    v2f a, b;
    a.x = h[k0 + lk]; a.y = h[k0 + lk + 1];
    b = brow2[(k0 + lk) >> 1];
    acc = wmma4(a, b, acc);
  }
  for (int k0 = 0; k0 < DDIM; k0 += 4) {
    v2f a, b;
    a.x = vr[k0 + lk]; a.y = vr[k0 + lk + 1];
    b = brow2[(DDIM + k0 + lk) >> 1];
    acc = wmma4(a, b, acc);
  }
  float bn = bias[n];
  for (int g = 0; g < 8; ++g) {
    int mm = mbase + g + ((lane >> 4) << 3);
    if (mm < NTOK) {
      float v = acc[g] + bn;
      if (!rv[mm / WWDIM]) v = 0.f;
      outb[(size_t)mm * DDIM + n] = v;
    }
  }
}

// ---------------- residual add + LayerNorm (warp per token row) ----------
__global__ void k_addln(const float* __restrict__ a, const float* __restrict__ b,
                        const float* __restrict__ g, const float* __restrict__ be,
                        float* __restrict__ out) {
  int warp = (int)(((size_t)blockIdx.x * blockDim.x + threadIdx.x) >> 5);
  int lane = threadIdx.x & 31;
  if (warp >= NTOK) return;
  size_t base = (size_t)warp * DDIM;
  float v[4];
  float s = 0.f;
  for (int q = 0; q < 4; ++q) {
    v[q] = a[base + lane + 32 * q] + b[base + lane + 32 * q];
    s += v[q];
  }
  for (int o = 16; o; o >>= 1) s += __shfl_xor(s, o);
  float mu = s * (1.f / 128.f);
  float ss = 0.f;
  for (int q = 0; q < 4; ++q) { float d = v[q] - mu; ss += d * d; }
  for (int o = 16; o; o >>= 1) ss += __shfl_xor(ss, o);
  float rstd = rsqrtf(ss * (1.f / 128.f) + 1e-5f);
  for (int q = 0; q < 4; ++q) {
    int c = lane + 32 * q;
    out[base + c] = (v[q] - mu) * rstd * g[c] + be[c];
  }
}

// ---------------- masked 3x3 conv (NHWC implicit GEMM), optional GELU ----
__global__ void k_conv(const float* __restrict__ in, const int* __restrict__ mask,
                       const float* __restrict__ wt, float* __restrict__ out,
                       int dogelu) {
  int wave = threadIdx.x >> 5, lane = threadIdx.x & 31;
  int lm = lane & 15, lk = (lane >> 4) * 2;
  int mbase = blockIdx.x * 16;
  int nbase = wave * 16;
  int m = mbase + lm;
  int mr = (m < NTOK) ? m : NTOK - 1;
  int b = mr / (LLDIM * WWDIM);
  int rem = mr % (LLDIM * WWDIM);
  int i = rem / WWDIM, j = rem % WWDIM;
  int n = nbase + lm;
  const float* wrow = wt + (size_t)n * (9 * DDIM);
  v8f acc = {};
  for (int tap = 0; tap < 9; ++tap) {
    int di = tap / 3 - 1, dj = tap % 3 - 1;
    int ii = i + di, jj = j + dj;
    bool ok = (ii >= 0 && ii < LLDIM && jj >= 0 && jj < WWDIM);
    int pix = ok ? ((b * LLDIM + ii) * WWDIM + jj) : 0;
    if (ok && mask[pix] != 0) ok = false;
    const float* arow = in + (size_t)pix * DDIM;
    const float* brow = wrow + tap * DDIM;
    for (int k0 = 0; k0 < DDIM; k0 += 4) {
      v2f a2, b2;
      a2.x = ok ? arow[k0 + lk] : 0.f;
      a2.y = ok ? arow[k0 + lk + 1] : 0.f;
      b2.x = brow[k0 + lk];
      b2.y = brow[k0 + lk + 1];
      acc = wmma4(a2, b2, acc);
    }
  }
  for (int g = 0; g < 8; ++g) {
    int mm = mbase + g + ((lane >> 4) << 3);
    if (mm < NTOK) {
      float v = acc[g];
      if (dogelu) v = 0.5f * v * (1.f + erff(v * 0.7071067811865475f));
      out[(size_t)mm * DDIM + n] = v;
    }
  }
}

extern "C" void kernel_launch(void* const* d_in, const int* in_sizes, int n_in,
                              void* d_out, int out_size, void* d_ws, size_t ws_size,
                              hipStream_t stream) {
  const float* x   = (const float*)d_in[0];
  const int*   msk = (const int*)d_in[1];
  const float* pe  = (const float*)d_in[2];
  const float* hqw = (const float*)d_in[3];
  const float* hqb = (const float*)d_in[4];
  const float* vqw = (const float*)d_in[5];
  const float* vqb = (const float*)d_in[6];
  const float* dw  = (const float*)d_in[7];
  const float* db  = (const float*)d_in[8];
  const float* l1g = (const float*)d_in[9];
  const float* l1b = (const float*)d_in[10];
  const float* c1w = (const float*)d_in[11];
  const float* c2w = (const float*)d_in[12];
  const float* l2g = (const float*)d_in[13];
  const float* l2b = (const float*)d_in[14];
  float* out = (float*)d_out;

  // workspace layout (floats), with aggressive reuse
  const size_t TOKD = (size_t)NTOK * DDIM;     // 16,646,400
  const size_t TOKQ = (size_t)NTOK * 384;      // 49,939,200
  float* ws  = (float*)d_ws;
  float* xp  = ws;               // x_proc
  float* qkv = xp + TOKD;        // qkv buffer (reused as dense_out + ln1 later)
  float* dns = qkv;              // dense output (qkv dead by then)
  float* ln1 = qkv + TOKD;       // LN1 output (inside qkv region)
  float* hv  = qkv + TOKQ;       // h-branch attn out; later conv1 output
  float* vv  = hv + TOKD;        // v-branch attn out; later conv2 output
  float* wt1 = vv + TOKD;        // repacked conv1 weights (147456)
  float* wt2 = wt1 + (size_t)DDIM * DDIM * 9;
  int*   rv  = (int*)(wt2 + (size_t)DDIM * DDIM * 9);

  (void)in_sizes; (void)n_in; (void)out_size; (void)ws_size;

  // large dynamic LDS: 256KB attn scores, 128KB dense weights
  hipFuncSetAttribute(reinterpret_cast<const void*>(k_attn),
                      hipFuncAttributeMaxDynamicSharedMemorySize, 262144);
  hipFuncSetAttribute(reinterpret_cast<const void*>(k_dense),
                      hipFuncAttributeMaxDynamicSharedMemorySize, 131072);

  dim3 b256(256);
  k_prep<<<dim3((unsigned)((TOKD + 255) / 256)), b256, 0, stream>>>(x, pe, xp);
  k_rowvalid<<<dim3((NROW * 32 + 255) / 256), b256, 0, stream>>>(msk, rv);
  k_repack<<<dim3((DDIM * DDIM * 9 + 255) / 256), b256, 0, stream>>>(c1w, wt1);
  k_repack<<<dim3((DDIM * DDIM * 9 + 255) / 256), b256, 0, stream>>>(c2w, wt2);

  // horizontal attention branch
  k_qkv<<<dim3(MTILES, 3), b256, 0, stream>>>(xp, hqw, hqb, qkv, 0);
  k_rotary<<<dim3((unsigned)(((size_t)NTOK * 64 + 255) / 256)), b256, 0, stream>>>(qkv);
  k_attn<<<dim3(NROW), dim3(512), 262144, stream>>>(qkv, msk, hv);

  // vertical attention branch (diagonal gather fused into the QKV GEMM)
  k_qkv<<<dim3(MTILES, 3), b256, 0, stream>>>(xp, vqw, vqb, qkv, 1);
  k_rotary<<<dim3((unsigned)(((size_t)NTOK * 64 + 255) / 256)), b256, 0, stream>>>(qkv);
  k_attn<<<dim3(NROW), dim3(512), 262144, stream>>>(qkv, msk, vv);

  // dense (TDM-staged weights in LDS) + residual + LN1
  k_dense<<<dim3(MTILES), b256, 131072, stream>>>(hv, vv, dw, db, rv, dns);
  k_addln<<<dim3((unsigned)(((size_t)NTOK * 32 + 255) / 256)), b256, 0, stream>>>(
      dns, xp, l1g, l1b, ln1);

  // masked conv1 + GELU, masked conv2, residual + LN2 -> out
  k_conv<<<dim3(MTILES), b256, 0, stream>>>(ln1, msk, wt1, hv, 1);
  k_conv<<<dim3(MTILES), b256, 0, stream>>>(hv, msk, wt2, vv, 0);
  k_addln<<<dim3((unsigned)(((size_t)NTOK * 32 + 255) / 256)), b256, 0, stream>>>(
      vv, ln1, l2g, l2b, out);
}